// SoPa_MLP_33466385170709
// MI455X (gfx1250) — compile-verified
//
#include <hip/hip_runtime.h>
#include <math.h>

// ---------------------------------------------------------------------------
// SoPa-MLP for MI455X (gfx1250).
// Dominant cost: h = einsum('bnd,edh->bneh') : M=8192, K=300, N'=22400
//   -> fp32 WMMA 16x16x4 (exact fp32 path), fused with +b1, leaky-relu,
//      dot(w2)+b2, sigmoid. Each wave owns 1 expert x 4 M-tiles so the w1
//      (B-matrix) fragment is loaded once per k-step and reused by 4 WMMAs
//      (vmem/wmma = 0.5, w1 L2 traffic cut 4x vs 1 tile/wave).
// Then a 512-step recurrence (parallel over 1600 (b,p) pairs) and a tiny MLP.
// ---------------------------------------------------------------------------

typedef __attribute__((ext_vector_type(2))) float v2f;
typedef __attribute__((ext_vector_type(8))) float v8f;

namespace cfg {
constexpr int B    = 16;
constexpr int N    = 512;
constexpr int DIN  = 300;            // K, multiple of 4 -> 75 WMMA k-steps
constexpr int P    = 100;
constexpr int L    = 7;
constexpr int NDIAG= 2;
constexpr int D    = P * NDIAG * L;  // 1400 "experts"
constexpr int H    = 16;
constexpr int MLPH = 256;
constexpr int MT   = 4;              // M-tiles (16 rows) per wave -> B-frag reuse
constexpr int ROWS = MT * 16;        // 64 rows staged in LDS per block
constexpr int EPB  = 8;              // experts (waves) per block
constexpr int LDSK = 302;            // padded row stride (bank-conflict free, 8B align)
}

// ds_swizzle group-of-32: new_lane = ((lane & 0x1f) | 0) ^ xor ; xor<16 keeps
// the two 16-lane halves (independent C rows) separate.
#define SWZ_ADD(v, imm) \
  ((v) + __int_as_float(__builtin_amdgcn_ds_swizzle(__float_as_int(v), (imm))))

// ---------------------------------------------------------------------------
// Kernel 1: fused GEMM. grid = (8192/64 M-supertiles, 1400/EPB expert groups),
// 256 threads. Block stages a 64-row x-tile in LDS (77 KB, dynamic) shared by
// 8 waves; each wave = 1 expert x 4 accumulator tiles.
// ---------------------------------------------------------------------------
__global__ __launch_bounds__(256) void sopa_fused_gemm(
    const float* __restrict__ batch,   // (B, DIN, N)
    const float* __restrict__ w1,      // (D, DIN, H)
    const float* __restrict__ b1,      // (D, H)
    const float* __restrict__ w2,      // (D, H)
    const float* __restrict__ b2,      // (D,)
    float* __restrict__ trans)         // (N, B, D) = sigmoid(lin), scan layout
{
  using namespace cfg;
  extern __shared__ float xs[];        // ROWS * LDSK floats

  const int mtile = blockIdx.x;        // 0..127 supertiles of 64 rows
  const int tid   = threadIdx.x;
  const int row0  = mtile * ROWS;      // flat (b,n) row base; 64 | 512 so b fixed
  const int b     = row0 >> 9;
  const int n0    = row0 & (N - 1);

  // Coalesced cooperative load: xs[m][k] = batch[b, k, n0+m].
  // t = k*64 + m -> 64 consecutive lanes hit 64 consecutive n (256B segments).
  const float* bp = batch + (size_t)b * DIN * N + n0;
  for (int t = tid; t < ROWS * DIN; t += 256) {
    int k = t >> 6;
    int m = t & 63;
    xs[m * LDSK + k] = bp[(size_t)k * N + m];
  }
  __syncthreads();

  const int wave  = tid >> 5;
  const int lane  = tid & 31;
  const int e     = blockIdx.y * EPB + wave;      // expert 0..1399
  const int hcol  = lane & 15;                    // B/C/D column (h index)
  const int khalf = (lane >> 4) << 1;             // 0 or 2 (A/B K sub-block)

  const float* w1e  = w1 + (size_t)e * DIN * H;
  const float* arow = xs + (lane & 15) * LDSK;    // A row, M = lane%16 (tile 0)

  v8f zero = {};
  v8f acc[MT];
  #pragma unroll
  for (int t = 0; t < MT; ++t) acc[t] = zero;

  #pragma unroll 5
  for (int k0 = 0; k0 < DIN; k0 += 4) {
    // B 4x16 fragment: lane = h column, VGPR0/1 = K rows (khalf, khalf+1)
    const float* wk = w1e + (size_t)(k0 + khalf) * H + hcol;
    v2f bb;
    bb.x = wk[0];
    bb.y = wk[H];
    // 4 A fragments (16x4 fp32, ISA layout) from LDS feed 4 WMMAs per B load.
    #pragma unroll
    for (int t = 0; t < MT; ++t) {
      v2f a = *(const v2f*)(arow + t * 16 * LDSK + k0 + khalf); // ds_load_b64
      acc[t] = __builtin_amdgcn_wmma_f32_16x16x4_f32(
          /*neg_a=*/false, a, /*neg_b=*/false, bb,
          /*c_mod=*/(short)0, acc[t], /*reuse_a=*/false, /*reuse_b=*/false);
    }
  }

  // Epilogue: h -> leaky_relu(h + b1) * w2, 16-lane XOR-swizzle reduction over
  // the h columns, +b2, sigmoid, scatter to scan layout.
  const float b1v = b1[e * H + hcol];
  const float w2v = w2[e * H + hcol];
  const float b2v = b2[e];
  const int halfbase = (lane >> 4) * 8;  // VGPR i = rows i (lanes 0-15) / 8+i

  #pragma unroll
  for (int t = 0; t < MT; ++t) {
    #pragma unroll
    for (int i = 0; i < 8; ++i) {
      float v = acc[t][i] + b1v;
      v = v > 0.f ? v : 0.1f * v;
      float r = v * w2v;
      r = SWZ_ADD(r, 0x201F);  // xor 8
      r = SWZ_ADD(r, 0x101F);  // xor 4
      r = SWZ_ADD(r, 0x081F);  // xor 2
      r = SWZ_ADD(r, 0x041F);  // xor 1
      if (hcol == 0) {
        int   n = n0 + t * 16 + halfbase + i;
        float s = 1.0f / (1.0f + __expf(-(r + b2v)));
        trans[((size_t)n * B + b) * D + e] = s;
      }
    }
  }
}

// ---------------------------------------------------------------------------
// Kernel 2: 512-step recurrence, one thread per (b, p) pair (1600 total).
// ---------------------------------------------------------------------------
__global__ __launch_bounds__(256) void sopa_scan(
    const float* __restrict__ trans,    // (N, B, D)
    const int*   __restrict__ mask,     // (B, N)
    const float* __restrict__ epsilon,  // (P, L-1)
    float* __restrict__ scores)         // (B, P)
{
  using namespace cfg;
  const int idx = blockIdx.x * 256 + threadIdx.x;
  if (idx >= B * P) return;
  const int b = idx / P;
  const int p = idx % P;

  float eps[L - 1];
  #pragma unroll
  for (int i = 0; i < L - 1; ++i)
    eps[i] = 1.0f / (1.0f + __expf(-epsilon[p * (L - 1) + i]));

  const int endi = 3 + p / 25;  // END_STATES: pattern length (4 + p/25) minus 1

  float hid[L] = {1.f, 0.f, 0.f, 0.f, 0.f, 0.f, 0.f};
  float sc = 0.f;

  for (int n = 0; n < N; ++n) {
    // 14 transitions for (n,b,p): base is 8B aligned (1400*4 and p*56 are).
    const v2f* tm2 =
        (const v2f*)(trans + ((size_t)n * B + b) * D + p * (NDIAG * L));
    float tv[NDIAG * L];
    #pragma unroll
    for (int q = 0; q < 7; ++q) { v2f x = tm2[q]; tv[2*q] = x.x; tv[2*q+1] = x.y; }
    const float* t0 = tv;       // diag 0
    const float* t1 = tv + L;   // diag 1
    const float m = (float)mask[b * N + n];

    float ae[L];
    ae[0] = hid[0];
    #pragma unroll
    for (int l = 1; l < L; ++l) ae[l] = hid[l] + hid[l - 1] * eps[l - 1];

    float nh[L];
    nh[0] = 1.0f + ae[0] * t0[0];                       // main[0]=1 (pad const)
    #pragma unroll
    for (int l = 1; l < L; ++l) nh[l] = ae[l - 1] * t1[l - 1] + ae[l] * t0[l];

    sc += m * nh[endi];
    #pragma unroll
    for (int l = 0; l < L; ++l) hid[l] = nh[l];
  }
  scores[b * P + p] = sc;
}

// ---------------------------------------------------------------------------
// Kernel 3: head MLP.  r = relu(scores @ mw0 + mb0) @ mw1 + mb1 -> sigmoid.
// ---------------------------------------------------------------------------
__global__ __launch_bounds__(256) void sopa_head(
    const float* __restrict__ scores,  // (B, P)
    const float* __restrict__ mw0,     // (P, MLPH)
    const float* __restrict__ mb0,     // (MLPH,)
    const float* __restrict__ mw1,     // (MLPH, 1)
    const float* __restrict__ mb1,     // (1,)
    float* __restrict__ out)           // (B,)
{
  using namespace cfg;
  __shared__ float red[MLPH];
  const int b = blockIdx.x;
  const int j = threadIdx.x;

  float acc = mb0[j];
  for (int k = 0; k < P; ++k)
    acc += scores[b * P + k] * mw0[k * MLPH + j];
  acc = acc > 0.f ? acc : 0.f;
  red[j] = acc * mw1[j];
  __syncthreads();

  for (int s = MLPH / 2; s > 0; s >>= 1) {
    if (j < s) red[j] += red[j + s];
    __syncthreads();
  }
  if (j == 0) out[b] = 1.0f / (1.0f + __expf(-(red[0] + mb1[0])));
}

// ---------------------------------------------------------------------------
extern "C" void kernel_launch(void* const* d_in, const int* in_sizes, int n_in,
                              void* d_out, int out_size, void* d_ws, size_t ws_size,
                              hipStream_t stream) {
  using namespace cfg;
  const float* batch   = (const float*)d_in[0];
  const int*   mask    = (const int*)  d_in[1];
  const float* w1      = (const float*)d_in[2];
  const float* b1      = (const float*)d_in[3];
  const float* w2      = (const float*)d_in[4];
  const float* b2      = (const float*)d_in[5];
  const float* epsilon = (const float*)d_in[6];
  const float* mw0     = (const float*)d_in[7];
  const float* mb0     = (const float*)d_in[8];
  const float* mw1     = (const float*)d_in[9];
  const float* mb1     = (const float*)d_in[10];
  float* out = (float*)d_out;

  float* trans  = (float*)d_ws;                 // N*B*D floats (~45.9 MB)
  float* scores = trans + (size_t)N * B * D;    // B*P floats

  dim3 g1((B * N) / ROWS, D / EPB);             // (128, 175)
  size_t smem = (size_t)ROWS * LDSK * sizeof(float);  // 77,312 B
  sopa_fused_gemm<<<g1, 256, smem, stream>>>(batch, w1, b1, w2, b2, trans);

  dim3 g2((B * P + 255) / 256);                 // 7 blocks
  sopa_scan<<<g2, 256, 0, stream>>>(trans, mask, epsilon, scores);

  sopa_head<<<dim3(B), 256, 0, stream>>>(scores, mw0, mb0, mw1, mb1, out);
}